// GCN1_73658689126522
// MI455X (gfx1250) — compile-verified
//
#include <hip/hip_runtime.h>
#include <math.h>

typedef float v2f __attribute__((ext_vector_type(2)));
typedef float v8f __attribute__((ext_vector_type(8)));

#define KS   5
#define F1c  32
#define F2c  64
#define FHc  128
#define NCLS 10

// ---------- helpers ----------
static __device__ __forceinline__ int enc_f(float f) {
  int i = __float_as_int(f);
  return i >= 0 ? i : (i ^ 0x7fffffff);   // monotone float->int encoding
}
static __device__ __forceinline__ float dec_f(int i) {
  return __int_as_float(i >= 0 ? i : (i ^ 0x7fffffff));
}
static __device__ __forceinline__ float eluf(float x) { return x > 0.f ? x : expm1f(x); }

static __device__ __forceinline__ void spline_basis(float u0, float u1, int* wi, float* bb) {
  // degree-1 open B-spline, dim=2, kernel_size=KS
  float v0 = u0 * (KS - 1), v1 = u1 * (KS - 1);
  float l0 = floorf(v0), l1 = floorf(v1);
  float f0 = v0 - l0, f1 = v1 - l1;
  int i0 = (int)l0, i1 = (int)l1;
  int a0 = min(max(i0, 0), KS - 1), a1 = min(max(i0 + 1, 0), KS - 1);
  int b0 = min(max(i1, 0), KS - 1), b1 = min(max(i1 + 1, 0), KS - 1);
  float p0 = 1.f - f0, p1 = f0, q0 = 1.f - f1, q1 = f1;
  wi[0] = a0 + KS * b0; bb[0] = p0 * q0;
  wi[1] = a0 + KS * b1; bb[1] = p0 * q1;
  wi[2] = a1 + KS * b0; bb[2] = p1 * q0;
  wi[3] = a1 + KS * b1; bb[3] = p1 * q1;
}

// ---------- layer 1 (F0=1 -> 32): scatter ----------
__global__ void k_l1_edges(const int* __restrict__ ei, const float* __restrict__ ea,
                           const float* __restrict__ x, const float* __restrict__ W1,
                           float* __restrict__ agg, float* __restrict__ deg, int E) {
  int e = blockIdx.x * blockDim.x + threadIdx.x;
  if (e >= E) return;
  int s = ei[e], d = ei[E + e];
  int wi[4]; float bb[4];
  spline_basis(ea[2 * e], ea[2 * e + 1], wi, bb);
  float xs = x[s];
  atomicAdd(&deg[d], 1.0f);
  const float* w0 = W1 + wi[0] * F1c;
  const float* w1 = W1 + wi[1] * F1c;
  const float* w2 = W1 + wi[2] * F1c;
  const float* w3 = W1 + wi[3] * F1c;
  float* ag = agg + (size_t)d * F1c;
#pragma unroll 8
  for (int o = 0; o < F1c; ++o) {
    float w = bb[0] * w0[o] + bb[1] * w1[o] + bb[2] * w2[o] + bb[3] * w3[o];
    atomicAdd(&ag[o], xs * w);
  }
}

__global__ void k_l1_node(const float* __restrict__ agg, const float* __restrict__ deg,
                          const float* __restrict__ x, const float* __restrict__ root1,
                          const float* __restrict__ b1, float* __restrict__ h, int N) {
  long long t = blockIdx.x * (long long)blockDim.x + threadIdx.x;
  if (t >= (long long)N * F1c) return;
  int n = (int)(t / F1c), o = (int)(t % F1c);
  float v = agg[t] / fmaxf(deg[n], 1.f) + x[n] * root1[o] + b1[o];
  h[t] = eluf(v);
}

// ---------- generic pooling pieces ----------
__global__ void k_pool_cnt_pos(const int* __restrict__ cl, const float* __restrict__ pos,
                               float* __restrict__ cnt, float* __restrict__ pos2, int N) {
  int n = blockIdx.x * blockDim.x + threadIdx.x;
  if (n >= N) return;
  int c = cl[n];
  if (c < 0 || c >= N) return;
  atomicAdd(&cnt[c], 1.f);
  if (pos2) {
    atomicAdd(&pos2[2 * c], pos[2 * n]);
    atomicAdd(&pos2[2 * c + 1], pos[2 * n + 1]);
  }
}

__global__ void k_seg_max(const int* __restrict__ cl, const float* __restrict__ h,
                          int* __restrict__ enc, int N, int F) {
  long long t = blockIdx.x * (long long)blockDim.x + threadIdx.x;
  if (t >= (long long)N * F) return;
  int n = (int)(t / F), o = (int)(t % F);
  int c = cl[n];
  if (c < 0 || c >= N) return;
  atomicMax(&enc[(size_t)c * F + o], enc_f(h[t]));
}

__global__ void k_seg_fin(const int* __restrict__ enc, const float* __restrict__ cnt,
                          float* __restrict__ outv, int N, int F) {
  long long t = blockIdx.x * (long long)blockDim.x + threadIdx.x;
  if (t >= (long long)N * F) return;
  int n = (int)(t / F);
  outv[t] = cnt[n] > 0.f ? dec_f(enc[t]) : 0.f;
}

__global__ void k_pos2_fin(float* __restrict__ pos2, const float* __restrict__ cnt, int N) {
  int t = blockIdx.x * blockDim.x + threadIdx.x;
  if (t >= 2 * N) return;
  pos2[t] = pos2[t] / fmaxf(cnt[t >> 1], 1.f);
}

// ---------- pooled-graph Cartesian attrs + spline precompute ----------
__global__ void k_cart(const int* __restrict__ ei2, const float* __restrict__ pos2,
                       float* __restrict__ cart, float* __restrict__ maxabs, int E2) {
  int e = blockIdx.x * blockDim.x + threadIdx.x;
  if (e >= E2) return;
  int r = ei2[e], c = ei2[E2 + e];
  float c0 = pos2[2 * r] - pos2[2 * c];
  float c1 = pos2[2 * r + 1] - pos2[2 * c + 1];
  cart[2 * e] = c0;
  cart[2 * e + 1] = c1;
  float m = fmaxf(fabsf(c0), fabsf(c1));
  atomicMax((int*)maxabs, __float_as_int(m));   // non-negative: int order == float order
}

__global__ void k_spline2(const float* __restrict__ cart, const float* __restrict__ maxabs,
                          int* __restrict__ wi2, float* __restrict__ bb2, int E2) {
  int e = blockIdx.x * blockDim.x + threadIdx.x;
  if (e >= E2) return;
  float inv = 0.5f / maxabs[0];
  float u0 = cart[2 * e] * inv + 0.5f;
  float u1 = cart[2 * e + 1] * inv + 0.5f;
  int wi[4]; float bb[4];
  spline_basis(u0, u1, wi, bb);
  ((int4*)wi2)[e] = make_int4(wi[0], wi[1], wi[2], wi[3]);
  ((float4*)bb2)[e] = make_float4(bb[0], bb[1], bb[2], bb[3]);
}

__global__ void k_deg2(const int* __restrict__ ei2, float* __restrict__ deg, int E2) {
  int e = blockIdx.x * blockDim.x + threadIdx.x;
  if (e >= E2) return;
  atomicAdd(&deg[ei2[E2 + e]], 1.f);
}

// ---------- WMMA f32 GEMM: C[N,64] = A[N,32] * B[32,64] ----------
// V_WMMA_F32_16X16X4_F32: A frag 2 VGPRs (lanes 0-15: K=0,1; lanes 16-31: K=2,3),
// B frag mirrored (lane = column), C/D 8 VGPRs (M = r + 8*(lane>=16), N = lane&15).
__global__ void k_gemm32x64(const float* __restrict__ A, const float* __restrict__ B,
                            float* __restrict__ C, int nRows) {
  int lane = threadIdx.x & 31;
  int wid = threadIdx.x >> 5;
  int row0 = (blockIdx.x * (blockDim.x >> 5) + wid) * 16;
  if (row0 >= nRows) return;               // wave-uniform; EXEC stays all-ones
  int m = lane & 15;
  int hi = lane >> 4;                      // 0 or 1 (which K pair)
  const float* arow = A + (size_t)(row0 + m) * F1c + hi * 2;
  v8f c0 = {0}, c1 = {0}, c2 = {0}, c3 = {0};
#pragma unroll
  for (int kk = 0; kk < 8; ++kk) {         // K = 32 in steps of 4
    v2f a;
    a.x = arow[4 * kk];
    a.y = arow[4 * kk + 1];
    const float* bp = B + (size_t)(4 * kk + hi * 2) * F2c;
    v2f b;
    b.x = bp[m + 0];  b.y = bp[F2c + m + 0];
    c0 = __builtin_amdgcn_wmma_f32_16x16x4_f32(false, a, false, b, (short)0, c0, false, false);
    b.x = bp[m + 16]; b.y = bp[F2c + m + 16];
    c1 = __builtin_amdgcn_wmma_f32_16x16x4_f32(false, a, false, b, (short)0, c1, false, false);
    b.x = bp[m + 32]; b.y = bp[F2c + m + 32];
    c2 = __builtin_amdgcn_wmma_f32_16x16x4_f32(false, a, false, b, (short)0, c2, false, false);
    b.x = bp[m + 48]; b.y = bp[F2c + m + 48];
    c3 = __builtin_amdgcn_wmma_f32_16x16x4_f32(false, a, false, b, (short)0, c3, false, false);
  }
  float* crow = C + (size_t)(row0 + hi * 8) * F2c + m;
#pragma unroll
  for (int r = 0; r < 8; ++r) {
    crow[(size_t)r * F2c + 0]  = c0[r];
    crow[(size_t)r * F2c + 16] = c1[r];
    crow[(size_t)r * F2c + 32] = c2[r];
    crow[(size_t)r * F2c + 48] = c3[r];
  }
}

// ---------- layer-2 edge accumulation for one kernel index k ----------
__global__ void k_edge_k(const int* __restrict__ ei2, const int* __restrict__ wi2,
                         const float* __restrict__ bb2, const float* __restrict__ xWk,
                         float* __restrict__ agg2, int kidx, int E2) {
  long long t = blockIdx.x * (long long)blockDim.x + threadIdx.x;
  int e = (int)(t >> 3);
  if (e >= E2) return;
  int sub = (int)(t & 7);
  int4 w = ((const int4*)wi2)[e];
  float4 b = ((const float4*)bb2)[e];
  float coef = 0.f;
  bool hit = false;
  if (w.x == kidx) { coef += b.x; hit = true; }
  if (w.y == kidx) { coef += b.y; hit = true; }
  if (w.z == kidx) { coef += b.z; hit = true; }
  if (w.w == kidx) { coef += b.w; hit = true; }
  if (!hit) return;
  int s = ei2[e], d = ei2[E2 + e];
  const float* src = xWk + (size_t)s * F2c + sub * 8;
  float* dst = agg2 + (size_t)d * F2c + sub * 8;
#pragma unroll
  for (int i = 0; i < 8; ++i) atomicAdd(&dst[i], coef * src[i]);
}

__global__ void k_l2_node(const float* __restrict__ agg2, const float* __restrict__ deg,
                          const float* __restrict__ rootOut, const float* __restrict__ b2,
                          float* __restrict__ h2, int N) {
  long long t = blockIdx.x * (long long)blockDim.x + threadIdx.x;
  if (t >= (long long)N * F2c) return;
  int n = (int)(t / F2c), o = (int)(t % F2c);
  float v = agg2[t] / fmaxf(deg[n], 1.f) + rootOut[t] + b2[o];
  h2[t] = eluf(v);
}

// ---------- global mean pool over graphs ----------
__global__ void k_gcnt(const int* __restrict__ b3, float* __restrict__ cntg, int N, int NG) {
  int n = blockIdx.x * blockDim.x + threadIdx.x;
  if (n >= N) return;
  int b = b3[n];
  if (b < 0 || b > NG) return;
  atomicAdd(&cntg[b], 1.f);
}

__global__ void k_gsum(const int* __restrict__ b3, const float* __restrict__ cnt2,
                       const int* __restrict__ x3enc, float* __restrict__ sg, int N, int NG) {
  long long t = blockIdx.x * (long long)blockDim.x + threadIdx.x;
  if (t >= (long long)N * F2c) return;
  int n = (int)(t / F2c), o = (int)(t % F2c);
  int b = b3[n];
  if (b < 0 || b > NG) return;
  float x3 = cnt2[n] > 0.f ? dec_f(x3enc[t]) : 0.f;
  atomicAdd(&sg[(size_t)b * F2c + o], x3);
}

// ---------- FC head + log_softmax: one block per graph ----------
__global__ void k_head(const float* __restrict__ sg, const float* __restrict__ cntg,
                       const float* __restrict__ fc1w, const float* __restrict__ fc1b,
                       const float* __restrict__ fc2w, const float* __restrict__ fc2b,
                       float* __restrict__ out, int NG) {
  __shared__ float xg[F2c];
  __shared__ float hfc[FHc];
  __shared__ float logits[NCLS];
  int g = blockIdx.x;
  int t = threadIdx.x;
  if (g >= NG) return;
  if (t < F2c) xg[t] = sg[(size_t)g * F2c + t] / fmaxf(cntg[g], 1.f);
  __syncthreads();
  if (t < FHc) {
    float acc = fc1b[t];
    for (int o = 0; o < F2c; ++o) acc += xg[o] * fc1w[o * FHc + t];
    hfc[t] = eluf(acc);
  }
  __syncthreads();
  if (t < NCLS) {
    float acc = fc2b[t];
    for (int i = 0; i < FHc; ++i) acc += hfc[i] * fc2w[i * NCLS + t];
    logits[t] = acc;
  }
  __syncthreads();
  if (t == 0) {
    float mx = logits[0];
    for (int c = 1; c < NCLS; ++c) mx = fmaxf(mx, logits[c]);
    float s = 0.f;
    for (int c = 0; c < NCLS; ++c) s += expf(logits[c] - mx);
    float lse = mx + logf(s);
    for (int c = 0; c < NCLS; ++c) out[(size_t)g * NCLS + c] = logits[c] - lse;
  }
}

// ---------- host orchestration ----------
static inline unsigned gsz(long long n, int b) { return (unsigned)((n + b - 1) / b); }

extern "C" void kernel_launch(void* const* d_in, const int* in_sizes, int n_in,
                              void* d_out, int out_size, void* d_ws, size_t ws_size,
                              hipStream_t stream) {
  (void)n_in; (void)ws_size;
  const float* x     = (const float*)d_in[0];
  const float* pos   = (const float*)d_in[1];
  const float* ea    = (const float*)d_in[2];
  const int*   ei    = (const int*)d_in[3];
  const int*   cl1   = (const int*)d_in[4];
  const int*   ei2   = (const int*)d_in[5];
  const int*   cl2   = (const int*)d_in[6];
  const int*   b3    = (const int*)d_in[7];
  const float* W1    = (const float*)d_in[9];
  const float* root1 = (const float*)d_in[10];
  const float* b1    = (const float*)d_in[11];
  const float* W2    = (const float*)d_in[12];
  const float* root2 = (const float*)d_in[13];
  const float* b2    = (const float*)d_in[14];
  const float* fc1w  = (const float*)d_in[15];
  const float* fc1b  = (const float*)d_in[16];
  const float* fc2w  = (const float*)d_in[17];
  const float* fc2b  = (const float*)d_in[18];
  float* out = (float*)d_out;

  const int N  = in_sizes[0];
  const int E  = in_sizes[3] / 2;
  const int E2 = in_sizes[5] / 2;
  const int NG = out_size / NCLS;

  // ---- carve workspace (256B aligned) ----
  char* base = (char*)d_ws;
  size_t off = 0;
  auto carve = [&](size_t bytes) -> void* {
    void* r = base + off;
    off = (off + bytes + 255) & ~(size_t)255;
    return r;
  };
  float* fCnt     = (float*)carve((size_t)(N + 1) * 4);        // deg1 -> cnt1 -> deg2 -> cnt2
  float* agg1_x2  = (float*)carve((size_t)N * F1c * 4);        // agg1, later x2
  float* hbuf     = (float*)carve((size_t)N * F1c * 4);        // h (layer-1 output)
  int*   enc32    = (int*)carve((size_t)N * F1c * 4);          // pool-1 encoded max
  float* pos2     = (float*)carve((size_t)N * 2 * 4);
  float* cart     = (float*)carve((size_t)E2 * 2 * 4);
  int*   wi2      = (int*)carve((size_t)E2 * 4 * 4);
  float* bb2      = (float*)carve((size_t)E2 * 4 * 4);
  float* xWk_h2   = (float*)carve((size_t)N * F2c * 4);        // per-k GEMM out, later h2
  float* root_x3e = (float*)carve((size_t)(N + 1) * F2c * 4);  // rootOut, later x3enc
  float* agg2     = (float*)carve((size_t)N * F2c * 4);
  float* sg       = (float*)carve((size_t)(NG + 1) * F2c * 4);
  float* cntg     = (float*)carve((size_t)(NG + 1) * 4);
  float* maxabs   = (float*)carve(256);

  const int B = 256;

  // ---- layer 1 ----
  hipMemsetAsync(fCnt, 0, (size_t)(N + 1) * 4, stream);
  hipMemsetAsync(agg1_x2, 0, (size_t)N * F1c * 4, stream);
  k_l1_edges<<<gsz(E, B), B, 0, stream>>>(ei, ea, x, W1, agg1_x2, fCnt, E);
  k_l1_node<<<gsz((long long)N * F1c, B), B, 0, stream>>>(agg1_x2, fCnt, x, root1, b1, hbuf, N);

  // ---- pool 1 (graclus max-pool + mean positions) ----
  hipMemsetAsync(fCnt, 0, (size_t)(N + 1) * 4, stream);
  hipMemsetAsync(enc32, 0x80, (size_t)N * F1c * 4, stream);  // < enc(-1) sentinel; h > -1 post-ELU
  hipMemsetAsync(pos2, 0, (size_t)N * 2 * 4, stream);
  k_pool_cnt_pos<<<gsz(N, B), B, 0, stream>>>(cl1, pos, fCnt, pos2, N);
  k_seg_max<<<gsz((long long)N * F1c, B), B, 0, stream>>>(cl1, hbuf, enc32, N, F1c);
  k_pos2_fin<<<gsz((long long)2 * N, B), B, 0, stream>>>(pos2, fCnt, N);
  k_seg_fin<<<gsz((long long)N * F1c, B), B, 0, stream>>>(enc32, fCnt, agg1_x2, N, F1c);  // -> x2

  // ---- pooled-graph edge attrs + basis ----
  hipMemsetAsync(maxabs, 0, 4, stream);
  k_cart<<<gsz(E2, B), B, 0, stream>>>(ei2, pos2, cart, maxabs, E2);
  k_spline2<<<gsz(E2, B), B, 0, stream>>>(cart, maxabs, wi2, bb2, E2);

  // ---- layer 2: 25 x (WMMA GEMM + gather/scatter) ----
  hipMemsetAsync(agg2, 0, (size_t)N * F2c * 4, stream);
  hipMemsetAsync(fCnt, 0, (size_t)(N + 1) * 4, stream);
  k_deg2<<<gsz(E2, B), B, 0, stream>>>(ei2, fCnt, E2);
  const int WPB = 4;                         // waves per block
  unsigned gemmGrid = gsz((long long)((N + 15) / 16), WPB);
  for (int k = 0; k < KS * KS; ++k) {
    k_gemm32x64<<<gemmGrid, WPB * 32, 0, stream>>>(agg1_x2, W2 + (size_t)k * F1c * F2c, xWk_h2, N);
    k_edge_k<<<gsz((long long)E2 * 8, B), B, 0, stream>>>(ei2, wi2, bb2, xWk_h2, agg2, k, E2);
  }
  k_gemm32x64<<<gemmGrid, WPB * 32, 0, stream>>>(agg1_x2, root2, root_x3e, N);  // rootOut
  k_l2_node<<<gsz((long long)N * F2c, B), B, 0, stream>>>(agg2, fCnt, root_x3e, b2, xWk_h2, N);  // -> h2

  // ---- pool 2 + global mean pool ----
  hipMemsetAsync(root_x3e, 0x80, (size_t)N * F2c * 4, stream);   // -> x3enc
  hipMemsetAsync(fCnt, 0, (size_t)(N + 1) * 4, stream);
  hipMemsetAsync(sg, 0, (size_t)(NG + 1) * F2c * 4, stream);
  hipMemsetAsync(cntg, 0, (size_t)(NG + 1) * 4, stream);
  k_pool_cnt_pos<<<gsz(N, B), B, 0, stream>>>(cl2, pos, fCnt, (float*)nullptr, N);
  k_seg_max<<<gsz((long long)N * F2c, B), B, 0, stream>>>(cl2, xWk_h2, (int*)root_x3e, N, F2c);
  k_gcnt<<<gsz(N, B), B, 0, stream>>>(b3, cntg, N, NG);
  k_gsum<<<gsz((long long)N * F2c, B), B, 0, stream>>>(b3, fCnt, (const int*)root_x3e, sg, N, NG);

  // ---- FC head ----
  k_head<<<NG, FHc, 0, stream>>>(sg, cntg, fc1w, fc1b, fc2w, fc2b, out, NG);
}